// DirectionalMamba3D_1589137900238
// MI455X (gfx1250) — compile-verified
//
#include <hip/hip_runtime.h>
#include <hip/hip_bf16.h>

typedef __attribute__((ext_vector_type(2))) float v2f;
typedef __attribute__((ext_vector_type(8))) float v8f;
typedef __attribute__((ext_vector_type(4))) unsigned v4u;
typedef __attribute__((ext_vector_type(8))) unsigned v8u;

#define NC 128      // model channels
#define NL 48       // axis length (D=H=W=48)
#define NDI 256     // d_inner
#define NDS 16      // d_state
#define NDTR 8      // dt rank
#define NBATCH 2
#define NAB 6       // 3 axes * 2 batches
#define VOL (48*48*48)

// ---- input pointer indices (setup_inputs flattening order) ----
#define IN_X 0
#define MB(a)   (1 + (a)*9)   // +0 in_proj_w +1 conv_w +2 conv_b +3 x_proj_w +4 dt_proj_w +5 dt_proj_b +6 A_log +7 D +8 out_proj_w
#define LN_G(a) (28 + (a)*2)
#define LN_B(a) (29 + (a)*2)
#define UP_W(a) (34 + (a)*2)
#define UP_B(a) (35 + (a)*2)
#define WG_W 40
#define WG_B 41
#define WSW  42
#define WSB  43
#define WMW  44
#define WMB  45
#define PROJ_W 46
#define PROJ_B 47
#define RES_SC 48

struct Params { const float* p[49]; };

// ---- workspace layout (floats) ----
static constexpr int OF_SEQ   = 0;
static constexpr int OF_SEQLN = OF_SEQ   + 3*NBATCH*NL*NC;
static constexpr int OF_XZ    = OF_SEQLN + 3*NBATCH*NL*NC;
static constexpr int OF_XS    = OF_XZ    + NAB*NL*512;
static constexpr int OF_DBL   = OF_XS    + NAB*NL*NDI;
static constexpr int OF_DTV   = OF_DBL   + NAB*NL*40;
static constexpr int OF_YBUF  = OF_DTV   + NAB*NL*NDI;
static constexpr int OF_TOK   = OF_YBUF  + NAB*NL*NDI;
static constexpr int OF_ZAX   = OF_TOK   + 3*NBATCH*NL*NC;
static constexpr int OF_GM    = OF_ZAX   + 3*NBATCH*NL*NC;
static constexpr int OF_CTX   = OF_GM    + 3*NBATCH*NC;
static constexpr int OF_YAX   = OF_CTX   + 3*NBATCH*NC;
static constexpr int OF_YP    = OF_YAX   + 3*NBATCH*NL*NC;
static constexpr int OF_MEAN  = OF_YP    + 3*NBATCH*NL*NC;
static constexpr int OF_INV   = OF_MEAN  + NBATCH*NC;

__device__ __forceinline__ float siluf(float x) { return x / (1.0f + __expf(-x)); }
__device__ __forceinline__ float sigmf(float x) { return 1.0f / (1.0f + __expf(-x)); }

// ============ TDM: DMA a rows x cols fp32 2-D panel from global into LDS offset 0 ============
// Builds the Tensor DMA Descriptor (D#) per CDNA5 ISA ch.8:
//  group0: [1:0]=count=1 | [63:32]=lds_addr | [120:64]=global_addr | [127:126]=type=2
//  group1: data_size=4B, tensor_dim0=cols, tensor_dim1=rows, tile_dim0=cols,
//          tile_dim1=rows, tensor_dim0_stride=cols (contiguous), no pad/iterate/multicast.
// Issue from one wave only (TDM ignores EXEC); completion via TENSORcnt.
__device__ __forceinline__ void tdm_load_panel_lds0(const float* gsrc, unsigned rows,
                                                    unsigned cols) {
  unsigned long long ga = (unsigned long long)(const void*)gsrc;
  v4u g0;
  g0.x = 1u;                                            // count=1, user descriptor
  g0.y = 0u;                                            // lds_addr = 0
  g0.z = (unsigned)(ga & 0xFFFFFFFFull);                // global_addr[31:0]
  g0.w = (unsigned)((ga >> 32) & 0x01FFFFFFull) | (2u << 30);  // addr[56:32] | type=2
  v8u g1;
  g1[0] = (2u << 16);                                   // data_size = 2 (4 bytes)
  g1[1] = (cols & 0xFFFFu) << 16;                       // tensor_dim0[15:0] at [63:48]
  g1[2] = ((cols >> 16) & 0xFFFFu) | ((rows & 0xFFFFu) << 16); // dim0 hi | dim1 lo
  g1[3] = ((rows >> 16) & 0xFFFFu) | ((cols & 0xFFFFu) << 16); // dim1 hi | tile_dim0
  g1[4] = (rows & 0xFFFFu);                             // tile_dim1 (tile_dim2 = 0)
  g1[5] = cols;                                         // tensor_dim0_stride[31:0]
  g1[6] = 0u;                                           // stride0 hi | dim1_stride lo
  g1[7] = 0u;                                           // dim1_stride hi
  asm volatile("tensor_load_to_lds %0, %1" :: "s"(g0), "s"(g1) : "memory");
}

// ============ generic fp32 WMMA tile: D(16x16) = A(MxK,row-major)*B(NxK,row-major)^T ============
__device__ __forceinline__ v8f wmma_dotT(const float* A, int lda, const float* B, int ldb,
                                         int m0, int n0, int K) {
  const int lane = threadIdx.x & 31;
  const int half = lane >> 4;
  const int ln   = lane & 15;
  v8f acc = {};
  const float* ap = A + (size_t)(m0 + ln) * lda + 2 * half;
  const float* bp = B + (size_t)(n0 + ln) * ldb + 2 * half;
  for (int k = 0; k < K; k += 4) {
    v2f av, bv;
    av.x = ap[k]; av.y = ap[k + 1];
    bv.x = bp[k]; bv.y = bp[k + 1];
    acc = __builtin_amdgcn_wmma_f32_16x16x4_f32(false, av, false, bv, (short)0, acc,
                                                false, false);
  }
  return acc;
}

__device__ __forceinline__ void store_tile(float* out, int ldo, int m0, int n0, v8f acc) {
  const int lane = threadIdx.x & 31;
  const int half = lane >> 4;
  const int ln   = lane & 15;
#pragma unroll
  for (int v = 0; v < 8; ++v)
    out[(size_t)(m0 + v + 8 * half) * ldo + n0 + ln] = acc[v];
}

// ============ K1: fused triple-axis mean reduction (streams x once, b128 loads) ============
__global__ __launch_bounds__(256) void k_reduce(Params prm, float* ws) {
  const int bc = blockIdx.x;            // 0..255 -> (b,c)
  const int b = bc >> 7, c = bc & 127;
  const float4* xp = (const float4*)(prm.p[IN_X] + (size_t)bc * VOL);
  __shared__ float sD[NL], sH[NL], sW[NL];
  if (threadIdx.x < NL) { sD[threadIdx.x] = 0.f; sH[threadIdx.x] = 0.f; sW[threadIdx.x] = 0.f; }
  __syncthreads();
  for (int i = threadIdx.x; i < VOL / 4; i += 256) {
    __builtin_prefetch((const char*)(xp + i) + 4096, 0, 1);
    float4 v = xp[i];
    int d  = i / 576;            // 12*48
    int r  = i - d * 576;
    int h  = r / 12;
    int w4 = (r - h * 12) * 4;
    float s = v.x + v.y + v.z + v.w;
    atomicAdd(&sD[d], s);
    atomicAdd(&sH[h], s);
    atomicAdd(&sW[w4 + 0], v.x);
    atomicAdd(&sW[w4 + 1], v.y);
    atomicAdd(&sW[w4 + 2], v.z);
    atomicAdd(&sW[w4 + 3], v.w);
  }
  __syncthreads();
  if (threadIdx.x < NL) {
    const float inv = 1.0f / 2304.0f;
    const int l = threadIdx.x;
    ws[OF_SEQ + ((0 * 2 + b) * NL + l) * NC + c] = sD[l] * inv;
    ws[OF_SEQ + ((1 * 2 + b) * NL + l) * NC + c] = sH[l] * inv;
    ws[OF_SEQ + ((2 * 2 + b) * NL + l) * NC + c] = sW[l] * inv;
  }
}

// ============ K2: layernorm over channels, per (axis,b,l) row ============
__global__ __launch_bounds__(128) void k_layernorm(Params prm, float* ws) {
  const int row = blockIdx.x;           // 0..287
  const int a = row / 96;
  const float* src = ws + OF_SEQ + (size_t)row * NC;
  float* dst = ws + OF_SEQLN + (size_t)row * NC;
  __shared__ float red[128];
  float v = src[threadIdx.x];
  red[threadIdx.x] = v;
  __syncthreads();
  for (int s = 64; s > 0; s >>= 1) {
    if (threadIdx.x < s) red[threadIdx.x] += red[threadIdx.x + s];
    __syncthreads();
  }
  float m = red[0] / NC;
  __syncthreads();
  red[threadIdx.x] = (v - m) * (v - m);
  __syncthreads();
  for (int s = 64; s > 0; s >>= 1) {
    if (threadIdx.x < s) red[threadIdx.x] += red[threadIdx.x + s];
    __syncthreads();
  }
  float var = red[0] / NC;
  dst[threadIdx.x] = (v - m) * rsqrtf(var + 1e-5f) * prm.p[LN_G(a)][threadIdx.x]
                   + prm.p[LN_B(a)][threadIdx.x];
}

// ============ K3: xz = seqln @ in_proj_w^T  (TDM-staged A panel; WMMA M=48,N=512,K=128) ====
__global__ __launch_bounds__(256) void k_inproj(Params prm, float* ws) {
  __shared__ float Ash[NL * NC];        // first/only shared array -> LDS offset 0 (TDM target)
  const int ab = blockIdx.x, a = ab >> 1;
  const float* Ag = ws + OF_SEQLN + (size_t)ab * NL * NC;
  const float* W = prm.p[MB(a) + 0];    // (512,128)
  float* xz = ws + OF_XZ + (size_t)ab * NL * 512;
  if (threadIdx.x < 32) {               // one wave issues the tensor DMA
    tdm_load_panel_lds0(Ag, NL, NC);
    __builtin_amdgcn_s_wait_tensorcnt(0);
  }
  __syncthreads();
  const int wid = threadIdx.x >> 5;
  for (int t = wid; t < 3 * 32; t += 8) {
    int mt = t % 3, nt = t / 3;
    v8f acc = wmma_dotT(Ash, NC, W, NC, mt * 16, nt * 16, NC);
    store_tile(xz, 512, mt * 16, nt * 16, acc);
  }
}

// ============ K4: causal depthwise conv (K=4) + silu -> xs; x_proj; dt_proj+softplus ============
__global__ __launch_bounds__(256) void k_convxproj(Params prm, float* ws) {
  const int ab = blockIdx.x, a = ab >> 1;
  const float* xz  = ws + OF_XZ  + (size_t)ab * NL * 512;
  float* xs  = ws + OF_XS  + (size_t)ab * NL * NDI;
  float* dbl = ws + OF_DBL + (size_t)ab * NL * 40;
  float* dtv = ws + OF_DTV + (size_t)ab * NL * NDI;
  const float* cw = prm.p[MB(a) + 1];   // (256,4)
  const float* cb = prm.p[MB(a) + 2];
  const float* xpw = prm.p[MB(a) + 3];  // (40,256)
  const float* dtw = prm.p[MB(a) + 4];  // (256,8)
  const float* dtb = prm.p[MB(a) + 5];
  const int i = threadIdx.x;            // channel (256)
  float c0 = cw[i * 4 + 0], c1 = cw[i * 4 + 1], c2 = cw[i * 4 + 2], c3 = cw[i * 4 + 3];
  for (int l = 0; l < NL; ++l) {
    float acc = cb[i];
    if (l >= 3) acc += c0 * xz[(l - 3) * 512 + i];
    if (l >= 2) acc += c1 * xz[(l - 2) * 512 + i];
    if (l >= 1) acc += c2 * xz[(l - 1) * 512 + i];
    acc += c3 * xz[l * 512 + i];
    xs[l * NDI + i] = siluf(acc);
  }
  __syncthreads();
  for (int id = i; id < NL * 40; id += 256) {
    int l = id / 40, j = id - l * 40;
    float acc = 0.f;
    for (int k = 0; k < NDI; ++k) acc += xs[l * NDI + k] * xpw[j * NDI + k];
    dbl[id] = acc;
  }
  __syncthreads();
  for (int l = 0; l < NL; ++l) {
    float acc = dtb[i];
#pragma unroll
    for (int r = 0; r < NDTR; ++r) acc += dbl[l * 40 + r] * dtw[i * NDTR + r];
    dtv[l * NDI + i] = log1pf(__expf(acc));   // softplus
  }
}

// ============ K5: selective scan (thread=channel) + z-gate, then out_proj via WMMA ============
__global__ __launch_bounds__(256) void k_scan(Params prm, float* ws) {
  const int ab = blockIdx.x, a = ab >> 1;
  const int d = threadIdx.x;            // inner channel
  const float* Alog = prm.p[MB(a) + 6];
  const float* Dp   = prm.p[MB(a) + 7];
  const float* Wout = prm.p[MB(a) + 8]; // (128,256)
  const float* xz  = ws + OF_XZ  + (size_t)ab * NL * 512;
  const float* xs  = ws + OF_XS  + (size_t)ab * NL * NDI;
  const float* dbl = ws + OF_DBL + (size_t)ab * NL * 40;
  const float* dtv = ws + OF_DTV + (size_t)ab * NL * NDI;
  float* yb = ws + OF_YBUF + (size_t)ab * NL * NDI;
  float As[NDS], h[NDS];
#pragma unroll
  for (int s = 0; s < NDS; ++s) { As[s] = -__expf(Alog[d * NDS + s]); h[s] = 0.f; }
  const float Dv = Dp[d];
  for (int l = 0; l < NL; ++l) {
    float dt = dtv[l * NDI + d];
    float xv = xs[l * NDI + d];
    float y = 0.f;
#pragma unroll
    for (int s = 0; s < NDS; ++s) {
      float dA = __expf(dt * As[s]);
      h[s] = dA * h[s] + dt * dbl[l * 40 + 8 + s] * xv;
      y += h[s] * dbl[l * 40 + 24 + s];
    }
    y += xv * Dv;
    yb[l * NDI + d] = y * siluf(xz[l * 512 + 256 + d]);
  }
  __syncthreads();
  // tok(48x128) = yb(48x256) @ Wout^T
  float* tok = ws + OF_TOK + (size_t)ab * NL * NC;
  const int wid = threadIdx.x >> 5;
  for (int t = wid; t < 3 * 8; t += 8) {
    int mt = t % 3, nt = t / 3;
    v8f acc = wmma_dotT(yb, NDI, Wout, NDI, mt * 16, nt * 16, NDI);
    store_tile(tok, NC, mt * 16, nt * 16, acc);
  }
}

// ============ K6: z = tok @ up_w^T + up_b  (WMMA 48x128x128) ============
__global__ __launch_bounds__(256) void k_up(Params prm, float* ws) {
  const int ab = blockIdx.x, a = ab >> 1;
  const float* tok = ws + OF_TOK + (size_t)ab * NL * NC;
  const float* W = prm.p[UP_W(a)];
  const float* bb = prm.p[UP_B(a)];
  float* z = ws + OF_ZAX + (size_t)ab * NL * NC;
  const int wid = threadIdx.x >> 5, lane = threadIdx.x & 31;
  const int half = lane >> 4, ln = lane & 15;
  for (int t = wid; t < 24; t += 8) {
    int mt = t % 3, nt = t / 3;
    v8f acc = wmma_dotT(tok, NC, W, NC, mt * 16, nt * 16, NC);
    float bias = bb[nt * 16 + ln];
#pragma unroll
    for (int v = 0; v < 8; ++v)
      z[(mt * 16 + v + 8 * half) * NC + nt * 16 + ln] = acc[v] + bias;
  }
}

// ============ K7: per-axis global mean of Z (volume mean == axis mean) ============
__global__ __launch_bounds__(128) void k_gmean(Params prm, float* ws) {
  const int ab = blockIdx.x;
  const float* z = ws + OF_ZAX + (size_t)ab * NL * NC;
  float s = 0.f;
  for (int l = 0; l < NL; ++l) s += z[l * NC + threadIdx.x];
  ws[OF_GM + ab * NC + threadIdx.x] = s / NL;
}

// ============ K8: ctx = Wg @ concat(gm[o1], gm[o2]) + Wg_b ============
__global__ __launch_bounds__(128) void k_ctx(Params prm, float* ws) {
  const int ab = blockIdx.x, a = ab >> 1, b = ab & 1;
  const int o1 = (a == 0) ? 1 : 0;
  const int o2 = (a == 2) ? 1 : 2;
  const float* g1 = ws + OF_GM + (o1 * 2 + b) * NC;
  const float* g2 = ws + OF_GM + (o2 * 2 + b) * NC;
  const float* Wg = prm.p[WG_W];        // (128,256)
  const int c = threadIdx.x;
  float acc = prm.p[WG_B][c];
  for (int k = 0; k < NC; ++k) acc += Wg[c * 256 + k] * g1[k];
  for (int k = 0; k < NC; ++k) acc += Wg[c * 256 + 128 + k] * g2[k];
  ws[OF_CTX + ab * NC + c] = acc;
}

// ============ K9: gate chain — TDM-staged Z; Ws GEMM+relu, Wm GEMM+sigmoid*Z, proj GEMM ====
__global__ __launch_bounds__(256) void k_gate(Params prm, float* ws) {
  __shared__ float zsh[NL * NC];        // first shared array -> LDS offset 0 (TDM target)
  __shared__ float rsh[NL * NC];        // relu(ctx + st)
  const int ab = blockIdx.x, a = ab >> 1;
  const float* zg = ws + OF_ZAX + (size_t)ab * NL * NC;
  const float* ctx = ws + OF_CTX + ab * NC;
  if (threadIdx.x < 32) {               // one wave issues the tensor DMA for the Z panel
    tdm_load_panel_lds0(zg, NL, NC);
    __builtin_amdgcn_s_wait_tensorcnt(0);
  }
  __syncthreads();
  const int wid = threadIdx.x >> 5, lane = threadIdx.x & 31;
  const int half = lane >> 4, ln = lane & 15;
  // stage 1: r = relu(ctx + Ws@z + Ws_b)
  for (int t = wid; t < 24; t += 8) {
    int mt = t % 3, nt = t / 3;
    v8f acc = wmma_dotT(zsh, NC, prm.p[WSW], NC, mt * 16, nt * 16, NC);
    float bias = prm.p[WSB][nt * 16 + ln] + ctx[nt * 16 + ln];
#pragma unroll
    for (int v = 0; v < 8; ++v) {
      float r = acc[v] + bias;
      rsh[(mt * 16 + v + 8 * half) * NC + nt * 16 + ln] = r > 0.f ? r : 0.f;
    }
  }
  __syncthreads();
  // stage 2: Y = sigmoid(Wm@r + Wm_b) * z
  float* Y = ws + OF_YAX + (size_t)ab * NL * NC;
  for (int t = wid; t < 24; t += 8) {
    int mt = t % 3, nt = t / 3;
    v8f acc = wmma_dotT(rsh, NC, prm.p[WMW], NC, mt * 16, nt * 16, NC);
    float bias = prm.p[WMB][nt * 16 + ln];
#pragma unroll
    for (int v = 0; v < 8; ++v) {
      int row = mt * 16 + v + 8 * half, col = nt * 16 + ln;
      Y[row * NC + col] = sigmf(acc[v] + bias) * zsh[row * NC + col];
    }
  }
  __syncthreads();
  // stage 3: yp = P_axis @ Y  (proj_w column block for this axis)
  float* yp = ws + OF_YP + (size_t)ab * NL * NC;
  for (int t = wid; t < 24; t += 8) {
    int mt = t % 3, nt = t / 3;
    v8f acc = wmma_dotT(Y, NC, prm.p[PROJ_W] + a * NC, 384, mt * 16, nt * 16, NC);
    store_tile(yp, NC, mt * 16, nt * 16, acc);
  }
}

// ============ K10: separable instance-norm stats ============
__global__ __launch_bounds__(128) void k_stats(Params prm, float* ws) {
  const int b = blockIdx.x, c = threadIdx.x;
  float mtot = 0.f, vtot = 0.f;
  for (int a = 0; a < 3; ++a) {
    const float* yp = ws + OF_YP + (size_t)((a * 2 + b) * NL) * NC;
    float s1 = 0.f, s2 = 0.f;
    for (int l = 0; l < NL; ++l) { float v = yp[l * NC + c]; s1 += v; s2 += v * v; }
    float m = s1 / NL;
    mtot += m;
    vtot += s2 / NL - m * m;             // per-axis variance; cross-covariances are zero
  }
  ws[OF_MEAN + b * NC + c] = mtot;
  ws[OF_INV + b * NC + c] = rsqrtf(vtot + 1e-5f);
}

// ============ K11: finalize stream: out = x + rs*inv*((ya'+yc'+ys') - mean) ============
__global__ __launch_bounds__(256) void k_finalize(Params prm, float* ws, float* out) {
  const int bc = blockIdx.x;
  const int b = bc >> 7, c = bc & 127;
  __shared__ __align__(16) float ad[NL], ch[NL], sw[NL];
  if (threadIdx.x < NL) {
    const int l = threadIdx.x;
    ad[l] = ws[OF_YP + ((0 * 2 + b) * NL + l) * NC + c];
    ch[l] = ws[OF_YP + ((1 * 2 + b) * NL + l) * NC + c];
    sw[l] = ws[OF_YP + ((2 * 2 + b) * NL + l) * NC + c];
  }
  __syncthreads();
  const float m  = ws[OF_MEAN + b * NC + c];
  const float sc = prm.p[RES_SC][0] * ws[OF_INV + b * NC + c];
  const float4* xp = (const float4*)(prm.p[IN_X] + (size_t)bc * VOL);
  float4* op = (float4*)(out + (size_t)bc * VOL);
  for (int i = threadIdx.x; i < VOL / 4; i += 256) {
    __builtin_prefetch((const char*)(xp + i) + 4096, 0, 1);
    int dh = i / 12;
    int wq = i - dh * 12;
    int d = dh / 48, h = dh - d * 48;
    float base = ad[d] + ch[h] - m;
    float4 swv = *(const float4*)&sw[wq * 4];
    float4 xv = xp[i];
    float4 o;
    o.x = xv.x + sc * (base + swv.x);
    o.y = xv.y + sc * (base + swv.y);
    o.z = xv.z + sc * (base + swv.z);
    o.w = xv.w + sc * (base + swv.w);
    op[i] = o;
  }
}

extern "C" void kernel_launch(void* const* d_in, const int* in_sizes, int n_in,
                              void* d_out, int out_size, void* d_ws, size_t ws_size,
                              hipStream_t stream) {
  Params prm;
  for (int i = 0; i < 49 && i < n_in; ++i) prm.p[i] = (const float*)d_in[i];
  float* ws = (float*)d_ws;
  float* out = (float*)d_out;

  k_reduce   <<<NBATCH * NC, 256, 0, stream>>>(prm, ws);
  k_layernorm<<<3 * NBATCH * NL, 128, 0, stream>>>(prm, ws);
  k_inproj   <<<NAB, 256, 0, stream>>>(prm, ws);
  k_convxproj<<<NAB, 256, 0, stream>>>(prm, ws);
  k_scan     <<<NAB, 256, 0, stream>>>(prm, ws);
  k_up       <<<NAB, 256, 0, stream>>>(prm, ws);
  k_gmean    <<<NAB, 128, 0, stream>>>(prm, ws);
  k_ctx      <<<NAB, 128, 0, stream>>>(prm, ws);
  k_gate     <<<NAB, 256, 0, stream>>>(prm, ws);
  k_stats    <<<NBATCH, 128, 0, stream>>>(prm, ws);
  k_finalize <<<NBATCH * NC, 256, 0, stream>>>(prm, ws, out);
}